// SlotAttention_13305808683341
// MI455X (gfx1250) — compile-verified
//
#include <hip/hip_runtime.h>
#include <math.h>

typedef __attribute__((ext_vector_type(2))) float v2f;
typedef __attribute__((ext_vector_type(8))) float v8f;
typedef int v4i __attribute__((vector_size(4 * sizeof(int))));  // matches builtin param

#if __has_builtin(__builtin_amdgcn_tanhf)
#define FAST_TANH(x) __builtin_amdgcn_tanhf(x)
#else
#define FAST_TANH(x) tanhf(x)
#endif

#if __has_builtin(__builtin_amdgcn_global_load_async_to_lds_b128) && \
    __has_builtin(__builtin_amdgcn_s_wait_asynccnt)
#define ASYNC_COPY 1
#else
#define ASYNC_COPY 0
#endif

#define AS1 __attribute__((address_space(1)))
#define AS3 __attribute__((address_space(3)))

constexpr int kNB = 4;    // batch
constexpr int kL  = 512;  // sequence
constexpr int kD  = 128;  // feature
constexpr int kH  = 128;  // hidden

constexpr int kChunk  = 32;            // q rows staged per chunk
constexpr int kNChunk = kL / kChunk;   // 16 chunks
constexpr int kPadQ   = 132;           // row stride for p/q tiles (132 % 64 = 4 -> conflict-free)
constexpr int kPadA   = 516;           // row stride for attn tile  (516 % 64 = 4 -> conflict-free)

// ---------------------------------------------------------------------------
// Kernel 1:  P = X @ W1[:, :D]^T,  Q = X @ W1[:, D:]^T   (exact fp32 WMMA)
// One wave per 16x16 tile; A-operand shared between the P and Q accumulators.
// ---------------------------------------------------------------------------
__global__ __launch_bounds__(256) void slotattn_pq_wmma(
    const float* __restrict__ X,   // (B*L, D)
    const float* __restrict__ W1,  // (H, 2D)
    float* __restrict__ P,         // (B*L, H)
    float* __restrict__ Q)         // (B*L, H)
{
  const int lane = threadIdx.x & 31;
  const int wave = threadIdx.x >> 5;
  const int tile = blockIdx.x * 8 + wave;  // 1024 tiles = 128 row-tiles x 8 h-tiles
  const int i0 = (tile >> 3) << 4;
  const int h0 = (tile & 7) << 4;
  const int m  = lane & 15;
  const int kh = lane >> 4;

  const float* xrow = X  + (size_t)(i0 + m) * kD;
  const float* wrow = W1 + (size_t)(h0 + m) * (2 * kD);

  v8f accP = {};
  v8f accQ = {};
#pragma unroll 4
  for (int k = 0; k < kD; k += 4) {
    v2f a, bP, bQ;
    a.x  = xrow[k + 2 * kh + 0];
    a.y  = xrow[k + 2 * kh + 1];
    bP.x = wrow[k + 2 * kh + 0];
    bP.y = wrow[k + 2 * kh + 1];
    bQ.x = wrow[kD + k + 2 * kh + 0];
    bQ.y = wrow[kD + k + 2 * kh + 1];
    accP = __builtin_amdgcn_wmma_f32_16x16x4_f32(false, a, false, bP,
                                                 (short)0, accP, false, false);
    accQ = __builtin_amdgcn_wmma_f32_16x16x4_f32(false, a, false, bQ,
                                                 (short)0, accQ, false, false);
  }
#pragma unroll
  for (int v = 0; v < 8; ++v) {
    const int row = i0 + v + 8 * kh;
    P[(size_t)row * kH + h0 + m] = accP[v];
    Q[(size_t)row * kH + h0 + m] = accQ[v];
  }
}

// ---------------------------------------------------------------------------
// Kernel 2 (fused): per (b, 16-row i-tile):
//   * stage q in 32-row chunks via async global->LDS (double buffered)
//   * scores[i,j] = sum_h W2[h]*tanh(p[i,h]+q[j,h])   (VALU cost center)
//   * row softmax in LDS
//   * context tile = attn @ X[b] via f32 WMMA (A operand from LDS)
//   * writes both halves of concat([X, context])
// ---------------------------------------------------------------------------
__global__ __launch_bounds__(256) void slotattn_fused(
    const float* __restrict__ X,    // (B*L, D)
    const float* __restrict__ P,    // (B*L, H)
    const float* __restrict__ Q,    // (B*L, H)
    const float* __restrict__ W2,   // (H)
    float* __restrict__ OUT)        // (B, L, 2D)
{
  __shared__ float p_s[16][kPadQ];
  __shared__ float w2_s[kH];
  __shared__ float q_s[2][kChunk][kPadQ];
  __shared__ float att_s[16][kPadA];
  __shared__ float red16[16][17];

  const int tid = threadIdx.x;
  const int b   = blockIdx.x >> 5;         // 4 batches
  const int i0  = (blockIdx.x & 31) << 4;  // 32 i-tiles of 16

  const float* Xb = X + (size_t)b * kL * kD;
  const float* Qb = Q + (size_t)b * kL * kH;

  // ---- load p tile + W2, and emit the concat() input half of the output ----
  if (tid < kH) w2_s[tid] = W2[tid];
#pragma unroll
  for (int r = 0; r < 8; ++r) {
    const int idx = tid + 256 * r;     // 0..2047
    const int ir  = idx >> 7;          // 0..15
    const int d   = idx & 127;
    p_s[ir][d] = P[((size_t)b * kL + i0 + ir) * kH + d];
    OUT[((size_t)b * kL + i0 + ir) * (2 * kD) + d] = Xb[(size_t)(i0 + ir) * kD + d];
  }

  // ---- q chunk staging helpers ----
  auto stage = [&](int c, int buf) {
    const float* gq = Qb + (size_t)c * kChunk * kH;
#pragma unroll
    for (int r = 0; r < 4; ++r) {
      const int idx = tid + 256 * r;   // 0..1023 transfers of 16B
      const int row = idx >> 5;        // 0..31
      const int sg  = (idx & 31) << 2; // float offset, 16B granules
#if ASYNC_COPY
      __builtin_amdgcn_global_load_async_to_lds_b128(
          (AS1 v4i*)(gq + (size_t)row * kH + sg),
          (AS3 v4i*)&q_s[buf][row][sg], 0, 0);
#else
      *(float4*)&q_s[buf][row][sg] = *(const float4*)(gq + (size_t)row * kH + sg);
#endif
    }
  };

  const int il16 = tid & 15;   // i_local for score phase
  const int j16  = tid >> 4;   // base j_local (0..15)

#if ASYNC_COPY
  stage(0, 0);
  stage(1, 1);
#endif

  for (int c = 0; c < kNChunk; ++c) {
    const int buf = c & 1;
#if ASYNC_COPY
    if (c + 1 < kNChunk) __builtin_amdgcn_s_wait_asynccnt(4);
    else                 __builtin_amdgcn_s_wait_asynccnt(0);
    __syncthreads();                        // chunk c visible to whole block
#else
    __syncthreads();
    stage(c, buf);
    __syncthreads();
#endif
    // ---- scores for this chunk: 16 i x 32 j, 2 j per thread ----
#pragma unroll
    for (int jj = 0; jj < 2; ++jj) {
      const int jl = j16 + 16 * jj;         // 0..31
      float acc = 0.f;
#pragma unroll 8
      for (int h = 0; h < kH; ++h)
        acc = fmaf(w2_s[h], FAST_TANH(p_s[il16][h] + q_s[buf][jl][h]), acc);
      att_s[il16][c * kChunk + jl] = acc;
    }
    __syncthreads();                        // done reading q_s[buf]
#if ASYNC_COPY
    if (c + 2 < kNChunk) stage(c + 2, buf); // refill the buffer just freed
#endif
  }
  __syncthreads();

  // ---- softmax over j for each of the 16 rows (16 threads per row) ----
  const int il = tid >> 4;   // row 0..15
  const int sl = tid & 15;
  float mx = -3.0e38f;
  for (int j = sl; j < kL; j += 16) mx = fmaxf(mx, att_s[il][j]);
  red16[il][sl] = mx;
  __syncthreads();
  if (sl < 8) red16[il][sl] = fmaxf(red16[il][sl], red16[il][sl + 8]);
  __syncthreads();
  if (sl < 4) red16[il][sl] = fmaxf(red16[il][sl], red16[il][sl + 4]);
  __syncthreads();
  if (sl < 2) red16[il][sl] = fmaxf(red16[il][sl], red16[il][sl + 2]);
  __syncthreads();
  if (sl < 1) red16[il][0] = fmaxf(red16[il][0], red16[il][1]);
  __syncthreads();
  mx = red16[il][0];
  __syncthreads();

  float sum = 0.f;
  for (int j = sl; j < kL; j += 16) {
    const float e = __expf(att_s[il][j] - mx);
    att_s[il][j] = e;
    sum += e;
  }
  red16[il][sl] = sum;
  __syncthreads();
  if (sl < 8) red16[il][sl] += red16[il][sl + 8];
  __syncthreads();
  if (sl < 4) red16[il][sl] += red16[il][sl + 4];
  __syncthreads();
  if (sl < 2) red16[il][sl] += red16[il][sl + 2];
  __syncthreads();
  if (sl < 1) red16[il][0] += red16[il][1];
  __syncthreads();
  const float inv = 1.0f / red16[il][0];
  for (int j = sl; j < kL; j += 16) att_s[il][j] *= inv;
  __syncthreads();

  // ---- context tile: (16 x 512) @ (512 x 128), one 16-wide d-tile per wave ----
  const int lane = tid & 31;
  const int wave = tid >> 5;          // 8 waves -> d0 = wave*16
  const int d0 = wave << 4;
  const int m  = lane & 15;
  const int kh = lane >> 4;

  v8f acc = {};
#pragma unroll 4
  for (int k = 0; k < kL; k += 4) {
    v2f a, bb;
    a.x  = att_s[m][k + 2 * kh + 0];  // A operand straight from LDS
    a.y  = att_s[m][k + 2 * kh + 1];
    bb.x = Xb[(size_t)(k + 2 * kh + 0) * kD + d0 + m];
    bb.y = Xb[(size_t)(k + 2 * kh + 1) * kD + d0 + m];
    acc = __builtin_amdgcn_wmma_f32_16x16x4_f32(false, a, false, bb,
                                                (short)0, acc, false, false);
  }
#pragma unroll
  for (int v = 0; v < 8; ++v) {
    const int i = i0 + v + 8 * kh;
    OUT[((size_t)b * kL + i) * (2 * kD) + kD + d0 + m] = acc[v];
  }
}

// ---------------------------------------------------------------------------
extern "C" void kernel_launch(void* const* d_in, const int* in_sizes, int n_in,
                              void* d_out, int out_size, void* d_ws, size_t ws_size,
                              hipStream_t stream) {
  const float* X  = (const float*)d_in[0];  // (4,512,128)
  const float* W1 = (const float*)d_in[1];  // (128,256)
  const float* W2 = (const float*)d_in[2];  // (1,128)
  float* OUT = (float*)d_out;               // (4,512,256)

  float* P = (float*)d_ws;                  // 2048*128
  float* Q = P + (size_t)kNB * kL * kH;     // 2048*128

  slotattn_pq_wmma<<<128, 256, 0, stream>>>(X, W1, P, Q);
  slotattn_fused<<<kNB * (kL / 16), 256, 0, stream>>>(X, P, Q, W2, OUT);
}